// ScenarioEncoderModel_55765855371410
// MI455X (gfx1250) — compile-verified
//
#include <hip/hip_runtime.h>
#include <hip/hip_bf16.h>
#include <math.h>

typedef float v2f __attribute__((ext_vector_type(2)));
typedef float v8f __attribute__((ext_vector_type(8)));

// ---------------------------------------------------------------------------
// WMMA GEMM: C[M x 128] = A[M x 128] @ B[128 x 128], f32 in / f32 accumulate.
// block = 128 threads = 4 waves; wave w computes the 16x16 tile at
// (blockIdx.x*16, (blockIdx.y*4+w)*16).  grid = (ceil(M/16), 2).
// Per V_WMMA_F32_16X16X4_F32:
//   A frag: lanes 0-15 hold K={k0,k0+1}, lanes 16-31 K={k0+2,k0+3}, row=lane&15
//   B frag: VGPR0 = row k0 (lanes 0-15) / k0+2 (lanes 16-31); VGPR1 = k0+1 / k0+3
//   C: VGPR j -> row j + 8*(lane>=16), col = lane&15
// A-rows are clamped on load (harmless garbage); C buffers are padded to a
// multiple of 16 rows so stores need no guard -> zero predication anywhere.
// ---------------------------------------------------------------------------
__global__ void wmma_gemm_f32_128(const float* __restrict__ A,
                                  const float* __restrict__ B,
                                  float* __restrict__ C, int M) {
  const int lane = threadIdx.x & 31;
  const int wave = threadIdx.x >> 5;  // 0..3
  const int hi   = lane >> 4;         // 0 or 1
  const int r    = lane & 15;
  const int m0   = blockIdx.x << 4;
  const int n0   = ((blockIdx.y << 2) + wave) << 4;
  int row = m0 + r;
  if (row >= M) row = M - 1;  // clamp: padded C rows absorb the garbage
  const float* __restrict__ Arow = A + (size_t)row * 128 + (hi << 1);
  const float* __restrict__ Bp   = B + (size_t)(hi << 1) * 128 + n0 + r;
  v8f c = {0.f, 0.f, 0.f, 0.f, 0.f, 0.f, 0.f, 0.f};
#pragma unroll 8
  for (int k0 = 0; k0 < 128; k0 += 4) {
    v2f a = *(const v2f*)(Arow + k0);          // global_load_b64
    v2f b;
    b.x = Bp[(size_t)k0 * 128];
    b.y = Bp[(size_t)k0 * 128 + 128];
    c = __builtin_amdgcn_wmma_f32_16x16x4_f32(false, a, false, b, (short)0, c,
                                              false, false);
  }
  float* __restrict__ Cp = C + (size_t)(m0 + (hi << 3)) * 128 + n0 + r;
#pragma unroll
  for (int j = 0; j < 8; ++j) Cp[(size_t)j * 128] = c[j];  // unguarded stores
}

// ---------------------------------------------------------------------------
// Elementwise / setup kernels
// ---------------------------------------------------------------------------
__global__ void zero_f32(float* p, long long n) {
  long long t = (long long)blockIdx.x * blockDim.x + threadIdx.x;
  if (t < n) p[t] = 0.f;
}
__global__ void zero_u32(unsigned* p, long long n) {
  long long t = (long long)blockIdx.x * blockDim.x + threadIdx.x;
  if (t < n) p[t] = 0u;
}
__global__ void copy_f32(const float* __restrict__ a, float* __restrict__ b, long long n) {
  long long t = (long long)blockIdx.x * blockDim.x + threadIdx.x;
  if (t < n) b[t] = a[t];
}

// vehicle embedding: feats = [v/10, a, len/5, w/2, yaw]; out = feats @ W(5x128) + b
__global__ void vehicle_embed(const float* __restrict__ vel, const float* __restrict__ acc,
                              const float* __restrict__ len, const float* __restrict__ wid,
                              const float* __restrict__ yaw, const float* __restrict__ W,
                              const float* __restrict__ b, float* __restrict__ out, int NV) {
  long long t = (long long)blockIdx.x * blockDim.x + threadIdx.x;
  if (t >= (long long)NV * 128) return;
  int n = (int)(t >> 7), d = (int)(t & 127);
  float f0 = vel[n] * 0.1f, f1 = acc[n], f2 = len[n] * 0.2f, f3 = wid[n] * 0.5f, f4 = yaw[n];
  out[t] = b[d] + f0 * W[d] + f1 * W[128 + d] + f2 * W[256 + d] + f3 * W[384 + d] +
           f4 * W[512 + d];
}

// v2v edge attributes (10 dims) with zero rows for self-loops
__global__ void build_ea_v2v(const float* __restrict__ ro, const float* __restrict__ rp,
                             const float* __restrict__ dist, const float* __restrict__ rv,
                             const float* __restrict__ ra, const float* __restrict__ ld,
                             float* __restrict__ ea, int EV, int NV) {
  int e = blockIdx.x * blockDim.x + threadIdx.x;
  if (e >= EV + NV) return;
  float f[10];
  if (e < EV) {
    float o = ro[e];
    f[0] = cosf(o); f[1] = sinf(o);
    f[2] = rp[2 * e] * 0.01f; f[3] = rp[2 * e + 1] * 0.01f;
    f[4] = 1.f - fminf(dist[e], 30.f) * (1.f / 30.f);
    f[5] = rv[2 * e] * 0.1f; f[6] = rv[2 * e + 1] * 0.1f;
    f[7] = ra[2 * e]; f[8] = ra[2 * e + 1];
    f[9] = 1.f - fminf(ld[e], 30.f) * (1.f / 30.f);
  } else {
#pragma unroll
    for (int i = 0; i < 10; ++i) f[i] = 0.f;
  }
#pragma unroll
  for (int i = 0; i < 10; ++i) ea[(size_t)e * 10 + i] = f[i];
}

// l2l edge attributes: copy 16-dim attrs, zero rows for self-loops
__global__ void build_ea_l2l(const float* __restrict__ src, float* __restrict__ ea,
                             int ELL, int NL) {
  long long t = (long long)blockIdx.x * blockDim.x + threadIdx.x;
  if (t >= (long long)(ELL + NL) * 16) return;
  long long e = t >> 4;
  ea[t] = (e < ELL) ? src[t] : 0.f;
}

// v2l edge attributes (5 dims)
__global__ void build_ea_v2l(const float* __restrict__ arel, const float* __restrict__ aabs,
                             const float* __restrict__ he, const float* __restrict__ le,
                             float* __restrict__ ea, int E) {
  int e = blockIdx.x * blockDim.x + threadIdx.x;
  if (e >= E) return;
  float h = he[e];
  ea[(size_t)e * 5 + 0] = arel[e];
  ea[(size_t)e * 5 + 1] = aabs[e] * 0.01f;
  ea[(size_t)e * 5 + 2] = cosf(h);
  ea[(size_t)e * 5 + 3] = sinf(h);
  ea[(size_t)e * 5 + 4] = le[e];
}

// edge indices with appended self loops: ei is (2,E) row-major
__global__ void build_loop_idx(const int* __restrict__ ei, int* __restrict__ src,
                               int* __restrict__ dst, int E, int N) {
  int t = blockIdx.x * blockDim.x + threadIdx.x;
  if (t >= E + N) return;
  if (t < E) { src[t] = ei[t]; dst[t] = ei[E + t]; }
  else       { src[t] = t - E; dst[t] = t - E; }
}

// ---------------------------------------------------------------------------
// Attention kernels.  Order-preserving uint encoding for float atomicMax:
//   key(x) = x>=0 ? bits|0x80000000 : ~bits   (monotone increasing)
// ---------------------------------------------------------------------------
template <int F>
__global__ void edge_logits(const int* __restrict__ src, const int* __restrict__ dstI,
                            const float* __restrict__ ea, const float* __restrict__ Wek,
                            const float* __restrict__ q, const float* __restrict__ k,
                            float* __restrict__ lg, unsigned* __restrict__ menc, int E) {
  long long t = (long long)blockIdx.x * blockDim.x + threadIdx.x;
  if (t >= (long long)E * 8) return;
  int e = (int)(t >> 3), h = (int)(t & 7);
  int s = src[e], d = dstI[e];
  const float4* __restrict__ qp4 = (const float4*)(q + (size_t)d * 128 + h * 16);
  const float4* __restrict__ kp4 = (const float4*)(k + (size_t)s * 128 + h * 16);
  float fv[F];
#pragma unroll
  for (int f = 0; f < F; ++f) fv[f] = ea[(size_t)e * F + f];
  float acc = 0.f;
#pragma unroll
  for (int jj = 0; jj < 4; ++jj) {
    float4 qv = qp4[jj], kv = kp4[jj];
    float4 ek = make_float4(0.f, 0.f, 0.f, 0.f);
#pragma unroll
    for (int f = 0; f < F; ++f) {
      float4 wv = *(const float4*)(Wek + f * 128 + h * 16 + jj * 4);
      ek.x += fv[f] * wv.x; ek.y += fv[f] * wv.y;
      ek.z += fv[f] * wv.z; ek.w += fv[f] * wv.w;
    }
    acc += qv.x * (kv.x + ek.x) + qv.y * (kv.y + ek.y) +
           qv.z * (kv.z + ek.z) + qv.w * (kv.w + ek.w);
  }
  float l = acc * 0.25f;  // 1/sqrt(DH=16)
  lg[t] = l;
  unsigned key = __float_as_uint(l);
  key = (key & 0x80000000u) ? ~key : (key | 0x80000000u);
  atomicMax(&menc[(size_t)d * 8 + h], key);
}

__global__ void edge_exp(const int* __restrict__ dstI, float* __restrict__ lg,
                         const unsigned* __restrict__ menc, float* __restrict__ ssum, int E) {
  long long t = (long long)blockIdx.x * blockDim.x + threadIdx.x;
  if (t >= (long long)E * 8) return;
  int e = (int)(t >> 3), h = (int)(t & 7);
  int d = dstI[e];
  unsigned key = menc[(size_t)d * 8 + h];
  unsigned bits = (key & 0x80000000u) ? (key & 0x7FFFFFFFu) : ~key;
  float m = __uint_as_float(bits);
  float ev = __expf(lg[t] - m);
  lg[t] = ev;
  atomicAdd(&ssum[(size_t)d * 8 + h], ev);
}

// 4 output columns per thread (float4 gathers); t over E*32
template <int F>
__global__ void edge_agg(const int* __restrict__ src, const int* __restrict__ dstI,
                         const float* __restrict__ ea, const float* __restrict__ Wev,
                         const float* __restrict__ v, const float* __restrict__ lg,
                         const float* __restrict__ ssum, float* __restrict__ agg, int E) {
  long long t = (long long)blockIdx.x * blockDim.x + threadIdx.x;
  if (t >= (long long)E * 32) return;
  int e = (int)(t >> 5), g = (int)(t & 31);
  int dcol = g << 2, h = g >> 2;
  int s = src[e], d = dstI[e];
  float alpha = lg[(size_t)e * 8 + h] / ssum[(size_t)d * 8 + h];
  float4 vv = *(const float4*)(v + (size_t)s * 128 + dcol);
  float4 evv = make_float4(0.f, 0.f, 0.f, 0.f);
#pragma unroll
  for (int f = 0; f < F; ++f) {
    float af = ea[(size_t)e * F + f];
    float4 wv = *(const float4*)(Wev + f * 128 + dcol);
    evv.x += af * wv.x; evv.y += af * wv.y; evv.z += af * wv.z; evv.w += af * wv.w;
  }
  float* ap = agg + (size_t)d * 128 + dcol;
  atomicAdd(ap + 0, alpha * (vv.x + evv.x));
  atomicAdd(ap + 1, alpha * (vv.y + evv.y));
  atomicAdd(ap + 2, alpha * (vv.z + evv.z));
  atomicAdd(ap + 3, alpha * (vv.w + evv.w));
}

__global__ void gelu_ip(float* p, long long n) {
  long long t = (long long)blockIdx.x * blockDim.x + threadIdx.x;
  if (t >= n) return;
  float x = p[t];
  float inner = 0.7978845608028654f * (x + 0.044715f * x * x * x);
  p[t] = 0.5f * x * (1.f + tanhf(inner));
}

__global__ void add_residual(const float* __restrict__ xd, const float* __restrict__ tmp,
                             const float* __restrict__ bo, float* __restrict__ out,
                             long long n) {
  long long t = (long long)blockIdx.x * blockDim.x + threadIdx.x;
  if (t >= n) return;
  out[t] = xd[t] + tmp[t] + bo[(int)(t & 127)];
}

// ---------------------------------------------------------------------------
// Host orchestration
// ---------------------------------------------------------------------------
static inline long long cdivll(long long a, long long b) { return (a + b - 1) / b; }

extern "C" void kernel_launch(void* const* d_in, const int* in_sizes, int n_in,
                              void* d_out, int out_size, void* d_ws, size_t ws_size,
                              hipStream_t stream) {
  (void)n_in; (void)ws_size;
  auto F32 = [&](int i) { return (const float*)d_in[i]; };
  auto I32 = [&](int i) { return (const int*)d_in[i]; };

  const int NV  = in_sizes[0];
  const int EV  = in_sizes[5];
  const int EVL = in_sizes[11];
  const int NL  = in_sizes[15] / 128;
  const int ELL = in_sizes[16] / 16;
  const int ELV = in_sizes[20] / 2;
  const int EVt = EV + NV;    // v2v with self loops
  const int ELt = ELL + NL;   // l2l with self loops
  const int NVp = (NV + 15) & ~15;  // padded row counts for GEMM outputs
  const int NLp = (NL + 15) & ~15;

  // ---- workspace carve (bump allocator, 256B aligned) ----
  char* w = (char*)d_ws;
  auto alloc = [&](size_t bytes) -> void* {
    void* p = (void*)w; w += (bytes + 255) & ~(size_t)255; return p;
  };
  float* xd_v[2] = {(float*)alloc((size_t)NV * 128 * 4), (float*)alloc((size_t)NV * 128 * 4)};
  float* xd_l[2] = {(float*)alloc((size_t)NL * 128 * 4), (float*)alloc((size_t)NL * 128 * 4)};
  float* q_v = (float*)alloc((size_t)NVp * 128 * 4);   // GEMM outputs: padded
  float* k_v = (float*)alloc((size_t)NVp * 128 * 4);
  float* v_v = (float*)alloc((size_t)NVp * 128 * 4);
  float* q_l = (float*)alloc((size_t)NLp * 128 * 4);
  float* k_l = (float*)alloc((size_t)NLp * 128 * 4);
  float* v_l = (float*)alloc((size_t)NLp * 128 * 4);
  float* ea_v2v = (float*)alloc((size_t)EVt * 10 * 4);
  float* ea_l2l = (float*)alloc((size_t)ELt * 16 * 4);
  float* ea_v2l = (float*)alloc((size_t)EVL * 5 * 4);
  int* s_v2v = (int*)alloc((size_t)EVt * 4);
  int* d_v2v = (int*)alloc((size_t)EVt * 4);
  int* s_l2l = (int*)alloc((size_t)ELt * 4);
  int* d_l2l = (int*)alloc((size_t)ELt * 4);
  float* lg_v2v = (float*)alloc((size_t)EVt * 8 * 4);
  float* lg_l2l = (float*)alloc((size_t)ELt * 8 * 4);
  float* lg_v2l = (float*)alloc((size_t)EVL * 8 * 4);
  float* lg_l2v = (float*)alloc((size_t)ELV * 8 * 4);
  unsigned* m_v = (unsigned*)alloc((size_t)NV * 8 * 4);
  float*    sm_v = (float*)alloc((size_t)NV * 8 * 4);
  unsigned* m_l = (unsigned*)alloc((size_t)NL * 8 * 4);
  float*    sm_l = (float*)alloc((size_t)NL * 8 * 4);
  float* agg_v = (float*)alloc((size_t)NV * 128 * 4);
  float* agg_l = (float*)alloc((size_t)NL * 128 * 4);
  float* tmp = (float*)alloc((size_t)NVp * 128 * 4);   // GEMM output: padded

  const int B = 256;
  auto gemm128 = [&](const float* A, const float* Bw, float* C, int M) {
    dim3 g((unsigned)((M + 15) / 16), 2);
    wmma_gemm_f32_128<<<g, 128, 0, stream>>>(A, Bw, C, M);
  };

  // ---- setup ----
  vehicle_embed<<<(unsigned)cdivll((long long)NV * 128, B), B, 0, stream>>>(
      F32(0), F32(1), F32(2), F32(3), F32(4), F32(57), F32(58), xd_v[0], NV);
  copy_f32<<<(unsigned)cdivll((long long)NL * 128, B), B, 0, stream>>>(
      F32(15), xd_l[0], (long long)NL * 128);
  build_ea_v2v<<<(unsigned)cdivll(EVt, B), B, 0, stream>>>(
      F32(5), F32(6), F32(7), F32(8), F32(9), F32(10), ea_v2v, EV, NV);
  build_ea_l2l<<<(unsigned)cdivll((long long)ELt * 16, B), B, 0, stream>>>(
      F32(16), ea_l2l, ELL, NL);
  build_ea_v2l<<<(unsigned)cdivll(EVL, B), B, 0, stream>>>(
      F32(11), F32(12), F32(13), F32(14), ea_v2l, EVL);
  build_loop_idx<<<(unsigned)cdivll(EVt, B), B, 0, stream>>>(I32(17), s_v2v, d_v2v, EV, NV);
  build_loop_idx<<<(unsigned)cdivll(ELt, B), B, 0, stream>>>(I32(18), s_l2l, d_l2l, ELL, NL);

  const int* s_v2l = I32(19);              // vehicle srcs
  const int* dI_v2l = I32(19) + EVL;       // lanelet dsts
  const int* s_l2v = I32(20);              // lanelet srcs
  const int* dI_l2v = I32(20) + ELV;       // vehicle dsts

  int cur = 0;
  for (int L = 0; L < 2; ++L) {
    const int pb = 21 + L * 18;            // param leaf base for this layer
    const float* Wek_l2l = F32(pb + 0); const float* Wek_l2v = F32(pb + 1);
    const float* Wek_v2l = F32(pb + 2); const float* Wek_v2v = F32(pb + 3);
    const float* Wev_l2l = F32(pb + 4); const float* Wev_l2v = F32(pb + 5);
    const float* Wev_v2l = F32(pb + 6); const float* Wev_v2v = F32(pb + 7);
    const float* Wk_l = F32(pb + 8);  const float* Wk_v = F32(pb + 9);
    const float* Wo_l = F32(pb + 10); const float* Wo_v = F32(pb + 11);
    const float* Wq_l = F32(pb + 12); const float* Wq_v = F32(pb + 13);
    const float* Wv_l = F32(pb + 14); const float* Wv_v = F32(pb + 15);
    const float* bo_l = F32(pb + 16); const float* bo_v = F32(pb + 17);

    // q/k/v projections (WMMA)
    gemm128(xd_v[cur], Wq_v, q_v, NV);
    gemm128(xd_v[cur], Wk_v, k_v, NV);
    gemm128(xd_v[cur], Wv_v, v_v, NV);
    gemm128(xd_l[cur], Wq_l, q_l, NL);
    gemm128(xd_l[cur], Wk_l, k_l, NL);
    gemm128(xd_l[cur], Wv_l, v_l, NL);

    // ===== dst = vehicle (incoming: v2v, l2v) =====
    zero_u32<<<(unsigned)cdivll((long long)NV * 8, B), B, 0, stream>>>(m_v, (long long)NV * 8);
    zero_f32<<<(unsigned)cdivll((long long)NV * 8, B), B, 0, stream>>>(sm_v, (long long)NV * 8);
    zero_f32<<<(unsigned)cdivll((long long)NV * 128, B), B, 0, stream>>>(agg_v, (long long)NV * 128);
    edge_logits<10><<<(unsigned)cdivll((long long)EVt * 8, B), B, 0, stream>>>(
        s_v2v, d_v2v, ea_v2v, Wek_v2v, q_v, k_v, lg_v2v, m_v, EVt);
    edge_logits<5><<<(unsigned)cdivll((long long)ELV * 8, B), B, 0, stream>>>(
        s_l2v, dI_l2v, ea_v2l, Wek_l2v, q_v, k_l, lg_l2v, m_v, ELV);
    edge_exp<<<(unsigned)cdivll((long long)EVt * 8, B), B, 0, stream>>>(d_v2v, lg_v2v, m_v, sm_v, EVt);
    edge_exp<<<(unsigned)cdivll((long long)ELV * 8, B), B, 0, stream>>>(dI_l2v, lg_l2v, m_v, sm_v, ELV);
    edge_agg<10><<<(unsigned)cdivll((long long)EVt * 32, B), B, 0, stream>>>(
        s_v2v, d_v2v, ea_v2v, Wev_v2v, v_v, lg_v2v, sm_v, agg_v, EVt);
    edge_agg<5><<<(unsigned)cdivll((long long)ELV * 32, B), B, 0, stream>>>(
        s_l2v, dI_l2v, ea_v2l, Wev_l2v, v_l, lg_l2v, sm_v, agg_v, ELV);
    gelu_ip<<<(unsigned)cdivll((long long)NV * 128, B), B, 0, stream>>>(agg_v, (long long)NV * 128);
    gemm128(agg_v, Wo_v, tmp, NV);
    add_residual<<<(unsigned)cdivll((long long)NV * 128, B), B, 0, stream>>>(
        xd_v[cur], tmp, bo_v, xd_v[1 - cur], (long long)NV * 128);

    // ===== dst = lanelet (incoming: l2l, v2l) =====
    zero_u32<<<(unsigned)cdivll((long long)NL * 8, B), B, 0, stream>>>(m_l, (long long)NL * 8);
    zero_f32<<<(unsigned)cdivll((long long)NL * 8, B), B, 0, stream>>>(sm_l, (long long)NL * 8);
    zero_f32<<<(unsigned)cdivll((long long)NL * 128, B), B, 0, stream>>>(agg_l, (long long)NL * 128);
    edge_logits<16><<<(unsigned)cdivll((long long)ELt * 8, B), B, 0, stream>>>(
        s_l2l, d_l2l, ea_l2l, Wek_l2l, q_l, k_l, lg_l2l, m_l, ELt);
    edge_logits<5><<<(unsigned)cdivll((long long)EVL * 8, B), B, 0, stream>>>(
        s_v2l, dI_v2l, ea_v2l, Wek_v2l, q_l, k_v, lg_v2l, m_l, EVL);
    edge_exp<<<(unsigned)cdivll((long long)ELt * 8, B), B, 0, stream>>>(d_l2l, lg_l2l, m_l, sm_l, ELt);
    edge_exp<<<(unsigned)cdivll((long long)EVL * 8, B), B, 0, stream>>>(dI_v2l, lg_v2l, m_l, sm_l, EVL);
    edge_agg<16><<<(unsigned)cdivll((long long)ELt * 32, B), B, 0, stream>>>(
        s_l2l, d_l2l, ea_l2l, Wev_l2l, v_l, lg_l2l, sm_l, agg_l, ELt);
    edge_agg<5><<<(unsigned)cdivll((long long)EVL * 32, B), B, 0, stream>>>(
        s_v2l, dI_v2l, ea_v2l, Wev_v2l, v_v, lg_v2l, sm_l, agg_l, EVL);
    gelu_ip<<<(unsigned)cdivll((long long)NL * 128, B), B, 0, stream>>>(agg_l, (long long)NL * 128);
    gemm128(agg_l, Wo_l, tmp, NL);
    add_residual<<<(unsigned)cdivll((long long)NL * 128, B), B, 0, stream>>>(
        xd_l[cur], tmp, bo_l, xd_l[1 - cur], (long long)NL * 128);

    cur = 1 - cur;
  }

  copy_f32<<<(unsigned)cdivll((long long)out_size, B), B, 0, stream>>>(
      xd_v[cur], (float*)d_out, (long long)out_size);
}